// SemiConv2d_15960098472407
// MI455X (gfx1250) — compile-verified
//
#include <hip/hip_runtime.h>
#include <hip/hip_bf16.h>

// Max-plus conv2d for MI455X (gfx1250).
// x: [8,64,64,64] f32, ker: [64,64,3,3] f32, out: [8,64,64,64] f32.
// out[b,o,h,w] = max_{c,ki,kj} x[b,c,h+ki-1,w+kj-1] + ker[o,c,ki,kj], pad=-inf.
//
// CDNA5 features: global_load_async_to_lds_b32/b128 (ASYNCcnt async copy),
// s_wait_asynccnt software pipelining over double-buffered LDS, wave32-aware
// blocking (per-wave-uniform o-group -> LDS broadcast reads).
// WMMA is intentionally absent: the tropical (max,+) semiring cannot be
// expressed on the (+,*) matrix pipes.

#define TILE 16          // 16x16 output pixels per workgroup
#define HALO 18          // TILE + 2
#define XSTRIDE 19       // padded row stride (bank-conflict-free for wave32)
#define CCHUNK 8         // channels per pipeline stage
#define NCHUNK 8         // 64 / CCHUNK
#define OHALF 32         // output channels per workgroup
#define XELEMS (CCHUNK*HALO*HALO)          // 2592 elements staged per chunk
#define XITER 11                           // ceil(2592/256)
#define KVEC (OHALF*CCHUNK*9/4)            // 576 b128 transfers per chunk
#define KITER 3                            // ceil(576/256)
// per-wave async instruction count per stage = XITER + KITER = 14

__device__ __forceinline__ unsigned lds_off(const void* p) {
  // Low 32 bits of a generic pointer into LDS == LDS byte address
  // (the shared aperture base occupies the high 32 bits on gfx1250).
  return (unsigned)(unsigned long long)p;
}

__device__ __forceinline__ void issue_stage(const float* __restrict__ x,
                                            const float* __restrict__ ker,
                                            int b, int ohalf, int h0, int w0,
                                            int c0, float* xbuf, float* kbuf,
                                            int tid) {
  // --- x halo tile: CCHUNK x 18 x 18, one b32 async load per element ---
#pragma unroll
  for (int it = 0; it < XITER; ++it) {
    int idx = it * 256 + tid;
    if (idx > XELEMS - 1) idx = XELEMS - 1;          // clamp: uniform issue count
    int cc  = idx / (HALO * HALO);
    int rem = idx - cc * (HALO * HALO);
    int rr  = rem / HALO;
    int col = rem - rr * HALO;
    int gh = h0 - 1 + rr;
    int gw = w0 - 1 + col;
    int ghc = gh < 0 ? 0 : (gh > 63 ? 63 : gh);      // clamp OOB to a valid
    int gwc = gw < 0 ? 0 : (gw > 63 ? 63 : gw);      // address; fixed up later
    unsigned goff = 4u * (unsigned)(((b * 64 + (c0 + cc)) * 64 + ghc) * 64 + gwc);
    unsigned loff = lds_off(&xbuf[cc * (HALO * XSTRIDE) + rr * XSTRIDE + col]);
    asm volatile("global_load_async_to_lds_b32 %0, %1, %2"
                 :: "v"(loff), "v"(goff), "s"(x) : "memory");
  }
  // --- weights: 32 o x (8 c x 9 taps) = 72 contiguous floats per o, b128 ---
#pragma unroll
  for (int it = 0; it < KITER; ++it) {
    int idx = it * 256 + tid;
    if (idx > KVEC - 1) idx = KVEC - 1;
    int o  = idx / 18;                               // 72 floats = 18 float4
    int t4 = idx - o * 18;
    unsigned goff = 4u * (unsigned)((ohalf * OHALF + o) * 576 + c0 * 9 + t4 * 4);
    unsigned loff = lds_off(&kbuf[o * 72 + t4 * 4]);
    asm volatile("global_load_async_to_lds_b128 %0, %1, %2"
                 :: "v"(loff), "v"(goff), "s"(ker) : "memory");
  }
}

__device__ __forceinline__ void fixup_oob(int h0, int w0, float* xbuf, int tid) {
  const float NI = -__builtin_inff();
#pragma unroll
  for (int it = 0; it < XITER; ++it) {
    int idx = it * 256 + tid;
    if (idx >= XELEMS) continue;
    int cc  = idx / (HALO * HALO);
    int rem = idx - cc * (HALO * HALO);
    int rr  = rem / HALO;
    int col = rem - rr * HALO;
    int gh = h0 - 1 + rr;
    int gw = w0 - 1 + col;
    if (gh < 0 || gh > 63 || gw < 0 || gw > 63)
      xbuf[cc * (HALO * XSTRIDE) + rr * XSTRIDE + col] = NI;
  }
}

__device__ __forceinline__ void compute_chunk(const float* __restrict__ xb,
                                              const float* __restrict__ kb,
                                              int r, int cg, int ogid,
                                              float acc[4][8]) {
#pragma unroll 1
  for (int cc = 0; cc < CCHUNK; ++cc) {
#pragma unroll
    for (int ki = 0; ki < 3; ++ki) {
      // 10-wide sliding x window for this (cc, ki); conflict-free stride 19.
      float xrow[10];
      const float* xp = &xb[cc * (HALO * XSTRIDE) + (r + ki) * XSTRIDE + cg * 8];
#pragma unroll
      for (int j = 0; j < 10; ++j) xrow[j] = xp[j];
#pragma unroll
      for (int kj = 0; kj < 3; ++kj) {
#pragma unroll
        for (int oo = 0; oo < 4; ++oo) {
          float kv = kb[(ogid * 4 + oo) * 72 + cc * 9 + ki * 3 + kj]; // broadcast
#pragma unroll
          for (int j = 0; j < 8; ++j)
            acc[oo][j] = fmaxf(acc[oo][j], xrow[j + kj] + kv);
        }
      }
    }
  }
}

__global__ __launch_bounds__(256) void maxplus_conv2d_kernel(
    const float* __restrict__ x, const float* __restrict__ ker,
    float* __restrict__ out) {
  __shared__ __align__(16) float sx[2][CCHUNK * HALO * XSTRIDE];  // 2x10944 B
  __shared__ __align__(16) float sk[2][OHALF * CCHUNK * 9];       // 2x 9216 B

  const int tid   = threadIdx.x;
  const int tx    = blockIdx.x & 3;
  const int ty    = blockIdx.x >> 2;
  const int b     = blockIdx.y;
  const int ohalf = blockIdx.z;
  const int h0 = ty * TILE, w0 = tx * TILE;

  const int ogid = tid >> 5;        // uniform per wave32 -> weight broadcast
  const int pg   = tid & 31;
  const int r    = pg >> 1;         // pixel row within tile
  const int cg   = pg & 1;          // pixel column group (8 wide)

  float acc[4][8];
  const float NI = -__builtin_inff();
#pragma unroll
  for (int oo = 0; oo < 4; ++oo)
#pragma unroll
    for (int j = 0; j < 8; ++j) acc[oo][j] = NI;

  const bool edge = (h0 == 0) || (h0 == 48) || (w0 == 0) || (w0 == 48);

  // ---- software pipeline: prime stage 0 ----
  issue_stage(x, ker, b, ohalf, h0, w0, 0, sx[0], sk[0], tid);

  for (int ch = 0; ch < NCHUNK; ++ch) {
    const int cur = ch & 1;
    if (ch + 1 < NCHUNK) {
      // Prefetch next chunk into the other buffer (last compute on that
      // buffer was chunk ch-1, barrier'd at the end of the previous iter).
      issue_stage(x, ker, b, ohalf, h0, w0, (ch + 1) * CCHUNK,
                  sx[cur ^ 1], sk[cur ^ 1], tid);
      // Wait only for stage ch: stage ch+1 (14 async ops) may stay in flight.
      asm volatile("s_wait_asynccnt 14" ::: "memory");
    } else {
      asm volatile("s_wait_asynccnt 0" ::: "memory");
    }
    __syncthreads();                     // all waves' stage-ch data landed
    if (edge) {
      fixup_oob(h0, w0, sx[cur], tid);   // -inf padding over clamped garbage
      __syncthreads();
    }
    compute_chunk(sx[cur], sk[cur], r, cg, ogid, acc);
    __syncthreads();                     // buffer safe for reuse
  }

  // ---- epilogue: 16B-aligned vector stores (global_store_b128) ----
  const int obase = ohalf * OHALF + ogid * 4;
#pragma unroll
  for (int oo = 0; oo < 4; ++oo) {
    float* op = out + (((size_t)(b * 64 + obase + oo) * 64 + (h0 + r)) * 64
                       + (w0 + cg * 8));
    float4 v0 = {acc[oo][0], acc[oo][1], acc[oo][2], acc[oo][3]};
    float4 v1 = {acc[oo][4], acc[oo][5], acc[oo][6], acc[oo][7]};
    *(float4*)(op)     = v0;
    *(float4*)(op + 4) = v1;
  }
}

extern "C" void kernel_launch(void* const* d_in, const int* in_sizes, int n_in,
                              void* d_out, int out_size, void* d_ws,
                              size_t ws_size, hipStream_t stream) {
  (void)in_sizes; (void)n_in; (void)d_ws; (void)ws_size; (void)out_size;
  const float* x   = (const float*)d_in[0];   // [8,64,64,64]
  const float* ker = (const float*)d_in[1];   // [64,64,3,3]
  float* out       = (float*)d_out;           // [8,64,64,64]
  dim3 grid(16, 8, 2);                        // 4x4 tiles, B=8, 2 o-halves
  dim3 block(256);                            // 8 wave32
  maxplus_conv2d_kernel<<<grid, block, 0, stream>>>(x, ker, out);
}